// RayleighKLLoss_mat_41790031790569
// MI455X (gfx1250) — compile-verified
//
#include <hip/hip_runtime.h>
#include <stdint.h>

// Problem constants (from reference): shapes (64, 4, 512, 512), 50 bins.
#define BATCH 64
#define NCH 4
#define HW (512 * 512)          // per-sample spatial elements = 262144
#define NBINS 50
#define EPSV 1e-8f
#define CLAMP_MINV 1e-6f

// Workspace layout (uint32 words):
//   [0 .. BATCH*2*NBINS)  histogram counts: per sample 50 p-bins then 50 t-bins
//   [MM_OFF .. MM_OFF+2B) per-sample {min,max} of pred magnitude, as float bits
#define CNT_WORDS (BATCH * 2 * NBINS)
#define MM_OFF CNT_WORDS

typedef unsigned int u32x4 __attribute__((ext_vector_type(4)));
typedef int i32x4 __attribute__((ext_vector_type(4)));
typedef int i32x8 __attribute__((ext_vector_type(8)));

typedef __attribute__((address_space(3))) const char lds_char_t;

// LDS byte offset of a __shared__ object (addrspacecast generic->AS3, ptr->int).
__device__ __forceinline__ unsigned lds_off(const void* p) {
  return (unsigned)(unsigned long long)(lds_char_t*)p;
}

// ---------------------------------------------------------------------------
// Tensor Data Mover: one 2D tile DMA (tile_dim0 elems x NCH channel rows,
// row stride = HW elements) global -> LDS.  D# per CDNA5 ISA ch.8.
// ---------------------------------------------------------------------------
__device__ __forceinline__ i32x8 make_g1_4ch_tile(int tile_dim0) {
  i32x8 g1;
  g1[0] = (2 << 16);                          // data_size = 2 (4 bytes)
  g1[1] = (int)((HW & 0xFFFF) << 16);         // tensor_dim0[15:0]
  g1[2] = (int)(((unsigned)HW >> 16) |        // tensor_dim0[31:16]
                ((NCH & 0xFFFF) << 16));      // tensor_dim1[15:0] = NCH
  g1[3] = (int)((tile_dim0 & 0xFFFF) << 16);  // tile_dim0
  g1[4] = NCH;                                // tile_dim1 = NCH, tile_dim2 = 0
  g1[5] = HW;                                 // tensor_dim0_stride[31:0]
  g1[6] = 0;
  g1[7] = 0;
  return g1;
}

__device__ __forceinline__ void tdm_load_tile(unsigned lds_byte,
                                              unsigned long long gaddr,
                                              i32x8 g1) {
  u32x4 g0;
  g0[0] = 1u;                                          // count=1 valid, user mode
  g0[1] = lds_byte;                                    // lds_addr
  g0[2] = (unsigned)(gaddr & 0xFFFFFFFFull);           // global_addr[31:0]
  g0[3] = (unsigned)((gaddr >> 32) & 0x01FFFFFFull)    // global_addr[56:32]
          | (2u << 30);                                // type = 2 ("image")
  i32x4 z4 = {0, 0, 0, 0};
#if defined(__clang_major__) && (__clang_major__ >= 23)
  i32x8 z8 = {0, 0, 0, 0, 0, 0, 0, 0};
  __builtin_amdgcn_tensor_load_to_lds(g0, g1, z4, z4, z8, 0);
#else
  __builtin_amdgcn_tensor_load_to_lds(g0, g1, z4, z4, 0);
#endif
}

// ---------------------------------------------------------------------------
// K2 (defined first so the disasm snippet shows its TDM prologue):
// histogram both streams.  Each wave runs a private TDM double-buffered
// pipeline: two DMAs per step (pred tile, true tile), TENSORcnt-throttled.
// ---------------------------------------------------------------------------
#define K2_TPB 128
#define K2_WAVES (K2_TPB / 32)             // 4
#define K2_BPS 64                          // blocks per sample
#define K2_CHUNK (HW / K2_BPS)             // 4096 elems per block
#define K2_PER_WAVE (K2_CHUNK / K2_WAVES)  // 1024 elems per wave
#define K2_TILE 128                        // elems per DMA tile (x NCH channels)
#define K2_STEPS (K2_PER_WAVE / K2_TILE)   // 8

__global__ __launch_bounds__(K2_TPB) void k_hist(const float* __restrict__ yp,
                                                 const float* __restrict__ yt,
                                                 unsigned* __restrict__ ws) {
  // [wave][dbuf][stream p/t][channel][elem] = 4*2*2*4*128 floats = 32 KB
  __shared__ float pipe[K2_WAVES][2][2][NCH][K2_TILE];
  __shared__ unsigned hist[2 * NBINS];

  const int tid = threadIdx.x;
  const int lane = tid & 31;
  const int wv = tid >> 5;
  const int wvs = __builtin_amdgcn_readfirstlane(wv);  // scalar for descriptors
  const int b = blockIdx.x / K2_BPS;
  const int blk = blockIdx.x % K2_BPS;

  if (tid < 2 * NBINS) hist[tid] = 0u;
  __syncthreads();

  const float mn = __uint_as_float(ws[MM_OFF + 2 * b + 0]);
  const float mx = __uint_as_float(ws[MM_OFF + 2 * b + 1]);
  const float w = (mx - mn) / (float)NBINS;

  const i32x8 g1 = make_g1_4ch_tile(K2_TILE);
  const unsigned long long baseP =
      (unsigned long long)(uintptr_t)(yp + (size_t)b * NCH * HW);
  const unsigned long long baseT =
      (unsigned long long)(uintptr_t)(yt + (size_t)b * NCH * HW);
  const int off0 = blk * K2_CHUNK + wvs * K2_PER_WAVE;  // channel-0 row offset

  const unsigned ldsBase = lds_off(&pipe[wvs][0][0][0][0]);
  const unsigned strideStream = NCH * K2_TILE * 4;  // 2048 B
  const unsigned strideBuf = 2 * strideStream;      // 4096 B

  // Prologue: step 0 into buffer 0.
  tdm_load_tile(ldsBase + 0 * strideStream, baseP + (unsigned long long)off0 * 4ull, g1);
  tdm_load_tile(ldsBase + 1 * strideStream, baseT + (unsigned long long)off0 * 4ull, g1);

  for (int s = 0; s < K2_STEPS; ++s) {
    const int buf = s & 1;
    if (s + 1 < K2_STEPS) {
      const unsigned nb = (unsigned)(buf ^ 1) * strideBuf;
      const unsigned long long goff =
          (unsigned long long)(off0 + (s + 1) * K2_TILE) * 4ull;
      tdm_load_tile(ldsBase + nb + 0 * strideStream, baseP + goff, g1);
      tdm_load_tile(ldsBase + nb + 1 * strideStream, baseT + goff, g1);
      __builtin_amdgcn_s_wait_tensorcnt(2);  // in-order: step s's two DMAs done
    } else {
      __builtin_amdgcn_s_wait_tensorcnt(0);
    }
    __asm__ __volatile__("" ::: "memory");  // DMA wrote LDS: force ds reloads

#pragma unroll
    for (int e = 0; e < K2_TILE / 32; ++e) {
      const int i = e * 32 + lane;
      const float p0 = pipe[wv][buf][0][0][i];
      const float p1 = pipe[wv][buf][0][1][i];
      const float p2 = pipe[wv][buf][0][2][i];
      const float p3 = pipe[wv][buf][0][3][i];
      const float t0 = pipe[wv][buf][1][0][i];
      const float t1 = pipe[wv][buf][1][1][i];
      const float t2 = pipe[wv][buf][1][2][i];
      const float t3 = pipe[wv][buf][1][3][i];
      float pm = fmaxf(sqrtf(p0 * p0 + p1 * p1 + p2 * p2 + p3 * p3), CLAMP_MINV);
      float tm = fmaxf(sqrtf(t0 * t0 + t1 * t1 + t2 * t2 + t3 * t3), CLAMP_MINV);
      int ip = (int)((pm - mn) / w);
      ip = min(max(ip, 0), NBINS - 1);
      atomicAdd(&hist[ip], 1u);
      if (tm >= mn && tm <= mx) {
        int it = (int)((tm - mn) / w);
        it = min(max(it, 0), NBINS - 1);
        atomicAdd(&hist[NBINS + it], 1u);
      }
    }
  }
  __syncthreads();
  if (tid < 2 * NBINS) {
    unsigned v = hist[tid];
    if (v) atomicAdd(&ws[b * 2 * NBINS + tid], v);
  }
}

// ---------------------------------------------------------------------------
// K1: per-sample min/max of pred magnitude, also via per-wave TDM pipeline.
// ---------------------------------------------------------------------------
#define K1_TPB 128
#define K1_WAVES (K1_TPB / 32)             // 4
#define K1_BPS 32                          // blocks per sample
#define K1_CHUNK (HW / K1_BPS)             // 8192 elems per block
#define K1_PER_WAVE (K1_CHUNK / K1_WAVES)  // 2048 elems per wave
#define K1_TILE 256                        // elems per DMA tile (x NCH channels)
#define K1_STEPS (K1_PER_WAVE / K1_TILE)   // 8

__global__ __launch_bounds__(K1_TPB) void k_minmax(const float* __restrict__ yp,
                                                   unsigned* __restrict__ ws) {
  // [wave][dbuf][channel][elem] = 4*2*4*256 floats = 32 KB
  __shared__ float pipe[K1_WAVES][2][NCH][K1_TILE];
  __shared__ float smn[K1_WAVES], smx[K1_WAVES];

  const int tid = threadIdx.x;
  const int lane = tid & 31;
  const int wv = tid >> 5;
  const int wvs = __builtin_amdgcn_readfirstlane(wv);
  const int b = blockIdx.x / K1_BPS;
  const int blk = blockIdx.x % K1_BPS;

  const i32x8 g1 = make_g1_4ch_tile(K1_TILE);
  const unsigned long long baseP =
      (unsigned long long)(uintptr_t)(yp + (size_t)b * NCH * HW);
  const int off0 = blk * K1_CHUNK + wvs * K1_PER_WAVE;

  const unsigned ldsBase = lds_off(&pipe[wvs][0][0][0]);
  const unsigned strideBuf = NCH * K1_TILE * 4;  // 4096 B

  float mn = __int_as_float(0x7F800000);
  float mx = 0.0f;

  // Prologue: step 0 into buffer 0.
  tdm_load_tile(ldsBase, baseP + (unsigned long long)off0 * 4ull, g1);

  for (int s = 0; s < K1_STEPS; ++s) {
    const int buf = s & 1;
    if (s + 1 < K1_STEPS) {
      const unsigned long long goff =
          (unsigned long long)(off0 + (s + 1) * K1_TILE) * 4ull;
      tdm_load_tile(ldsBase + (unsigned)(buf ^ 1) * strideBuf, baseP + goff, g1);
      __builtin_amdgcn_s_wait_tensorcnt(1);  // in-order: step s's DMA done
    } else {
      __builtin_amdgcn_s_wait_tensorcnt(0);
    }
    __asm__ __volatile__("" ::: "memory");

#pragma unroll
    for (int e = 0; e < K1_TILE / 32; ++e) {
      const int i = e * 32 + lane;
      const float a0 = pipe[wv][buf][0][i];
      const float a1 = pipe[wv][buf][1][i];
      const float a2 = pipe[wv][buf][2][i];
      const float a3 = pipe[wv][buf][3][i];
      float m = sqrtf(a0 * a0 + a1 * a1 + a2 * a2 + a3 * a3);
      m = fmaxf(m, CLAMP_MINV);
      mn = fminf(mn, m);
      mx = fmaxf(mx, m);
    }
  }

  // wave32 reduction, then cross-wave via LDS, then one global atomic pair.
  for (int o = 16; o > 0; o >>= 1) {
    mn = fminf(mn, __shfl_xor(mn, o, 32));
    mx = fmaxf(mx, __shfl_xor(mx, o, 32));
  }
  if (lane == 0) { smn[wv] = mn; smx[wv] = mx; }
  __syncthreads();
  if (tid == 0) {
    for (int i = 1; i < K1_WAVES; ++i) {
      mn = fminf(mn, smn[i]);
      mx = fmaxf(mx, smx[i]);
    }
    // non-negative floats: uint order == float order
    atomicMin(&ws[MM_OFF + 2 * b + 0], __float_as_uint(mn));
    atomicMax(&ws[MM_OFF + 2 * b + 1], __float_as_uint(mx));
  }
}

// ---------------------------------------------------------------------------
// K0: initialize workspace each launch (harness does not re-poison).
// ---------------------------------------------------------------------------
__global__ void k_init(unsigned* __restrict__ ws) {
  int i = blockIdx.x * blockDim.x + threadIdx.x;
  if (i < CNT_WORDS) ws[i] = 0u;
  if (i < BATCH) {
    ws[MM_OFF + 2 * i + 0] = 0x7F800000u;  // +inf  (min slot)
    ws[MM_OFF + 2 * i + 1] = 0u;           // 0.0f  (max slot; mags > 0)
  }
}

// ---------------------------------------------------------------------------
// K3: per-sample density-normalized KL, mean over batch -> d_out[0].
// ---------------------------------------------------------------------------
__global__ void k_kl(const unsigned* __restrict__ ws, float* __restrict__ out) {
  __shared__ float part[BATCH];
  const int b = threadIdx.x;
  if (b < BATCH) {
    const float mn = __uint_as_float(ws[MM_OFF + 2 * b + 0]);
    const float mx = __uint_as_float(ws[MM_OFF + 2 * b + 1]);
    const float w = (mx - mn) / (float)NBINS;
    const unsigned* cp = ws + (size_t)b * 2 * NBINS;
    const unsigned* ct = cp + NBINS;
    float sumT = 0.0f;
    for (int i = 0; i < NBINS; ++i) sumT += (float)ct[i];
    const float sp = (float)HW * w;  // c_p.sum * bin_width
    const float st = sumT * w;       // c_t.sum * bin_width
    float zp = 0.0f, zt = 0.0f;
    for (int i = 0; i < NBINS; ++i) {
      zp += (float)cp[i] / sp + EPSV;
      zt += (float)ct[i] / st + EPSV;
    }
    float kl = 0.0f;
    for (int i = 0; i < NBINS; ++i) {
      const float hp = ((float)cp[i] / sp + EPSV) / zp;
      const float ht = ((float)ct[i] / st + EPSV) / zt;
      kl += ht * logf(ht / hp);
    }
    part[b] = kl;
  }
  __syncthreads();
  if (threadIdx.x == 0) {
    float s = 0.0f;
    for (int i = 0; i < BATCH; ++i) s += part[i];
    out[0] = s / (float)BATCH;
  }
}

// ---------------------------------------------------------------------------
extern "C" void kernel_launch(void* const* d_in, const int* in_sizes, int n_in,
                              void* d_out, int out_size, void* d_ws, size_t ws_size,
                              hipStream_t stream) {
  (void)in_sizes; (void)n_in; (void)out_size; (void)ws_size;
  const float* yp = (const float*)d_in[0];
  const float* yt = (const float*)d_in[1];
  unsigned* ws = (unsigned*)d_ws;
  float* out = (float*)d_out;

  k_init<<<(CNT_WORDS + 255) / 256, 256, 0, stream>>>(ws);
  k_minmax<<<BATCH * K1_BPS, K1_TPB, 0, stream>>>(yp, ws);
  k_hist<<<BATCH * K2_BPS, K2_TPB, 0, stream>>>(yp, yt, ws);
  k_kl<<<1, BATCH, 0, stream>>>(ws, out);
}